// SupConLossWithQueue_8598524526701
// MI455X (gfx1250) — compile-verified
//
#include <hip/hip_runtime.h>
#include <math.h>

// Problem constants (from the reference)
#define QN     4096
#define BSZN   256
#define DN     128
#define BBN    4352      // B = Q + bsz
#define TWOB   8704      // 2B
#define NTILE  (TWOB / 16)   // 544 column tiles
#define SSTART 3841      // first selected row (s = Q - bsz + 1)
#define NROWS  4352      // number of selected rows (3841..8192)
#define DIAGTH 3840      // (g mod B) < 3840 -> diagonal self_mask value 1, else 0
#define INV_T  (1.0f/0.07f)

typedef __attribute__((ext_vector_type(2))) float v2f;
typedef __attribute__((ext_vector_type(8))) float v8f;

// contrast row c (0..8703) -> pointer into the original inputs.
// batch1 = [f2(0:256) | queue(256:4096) | f1], batch2 = [f1(0:256) | queue2(256:4096) | f2]
// f1 = features_flat rows 0..255, f2 = features_flat rows 256..511
__device__ __forceinline__ const float* contrast_row(int c, const float* __restrict__ feat,
                                                     const float* __restrict__ fq,
                                                     const float* __restrict__ fq2) {
  int half = (c >= BBN) ? 1 : 0;
  int c2 = half ? c - BBN : c;
  if (c2 >= QN) {
    int t = c2 - QN;                       // 0..255
    return feat + (size_t)(half ? (BSZN + t) : t) * DN;   // b1 tail=f1, b2 tail=f2
  } else if (c2 < BSZN) {
    return feat + (size_t)(half ? c2 : (BSZN + c2)) * DN; // b1 head=f2, b2 head=f1
  } else {
    return (half ? fq2 : fq) + (size_t)c2 * DN;
  }
}

__global__ void build_labels_kernel(const int* __restrict__ label,
                                    const int* __restrict__ lqueue,
                                    int* __restrict__ labels_full) {
  int j = blockIdx.x * 256 + threadIdx.x;
  if (j >= TWOB) return;
  int c2 = (j >= BBN) ? j - BBN : j;
  int v;
  if (c2 >= QN)        v = label[c2 - QN];
  else if (c2 < BSZN)  v = label[c2];      // lq[:256] = label
  else                 v = lqueue[c2];
  labels_full[j] = v;
}

// One workgroup = 8 waves * 16 rows = 128 selected rows.
// Double-buffered 16x128 column tiles staged with GLOBAL_LOAD_ASYNC_TO_LDS_B128
// (ASYNCcnt), overlapping the next tile's DMA with the current tile's WMMA chain.
// Pass 1: online (max, sum exp*logits_mask) over all 8704 columns via fp32 WMMA.
// Pass 2: recompute tiles, accumulate mask_f * min(0, x - lse) and mask_f.
__global__ __launch_bounds__(256)
void supcon_main_kernel(const float* __restrict__ feat,
                        const float* __restrict__ fq,
                        const float* __restrict__ fq2,
                        const int*   __restrict__ labels_full,
                        float*       __restrict__ rowres) {
  __shared__ float Bt[2][16 * DN];         // two 8 KB column-tile buffers

  const int tid  = threadIdx.x;
  const int lane = tid & 31;
  const int wave = tid >> 5;
  const int hi   = lane >> 4;              // half-wave select
  const int lo   = lane & 15;

  const int rowblock = blockIdx.x * 128 + wave * 16;   // first selected row of this wave

  // Per-thread slice of the cooperative tile load: row tid>>4, 32 bytes at seg.
  const int ldrow = tid >> 4;
  const int ldseg = (tid & 15) * 8;

  // Issue async global->LDS copy of column tile starting at jt into buffer buf.
  auto issue_tile = [&](int jt, int buf) {
    const float* bp = contrast_row(jt + ldrow, feat, fq, fq2) + ldseg;
    unsigned la = (unsigned)(uintptr_t)&Bt[buf][ldrow * DN + ldseg];  // LDS byte offset
    unsigned long long ga = (unsigned long long)(uintptr_t)bp;
    asm volatile(
        "global_load_async_to_lds_b128 %0, %1, off\n\t"
        "global_load_async_to_lds_b128 %0, %1, off offset:16"
        :: "v"(la), "v"(ga) : "memory");
  };

  // ---- A tile (16 rows x 128) into registers: one float2 per lane per k-step ----
  const int gA = SSTART + rowblock + lo;               // contrast row for this lane's A row
  const float* ap = contrast_row(gA, feat, fq, fq2);
  v2f areg[32];
#pragma unroll
  for (int kk = 0; kk < 32; ++kk)
    areg[kk] = *(const v2f*)(ap + 4 * kk + 2 * hi);

  // ---- per-lane row metadata: rows v + 8*hi of the tile ----
  int   rlab[8];
  float dv[8];
#pragma unroll
  for (int v = 0; v < 8; ++v) {
    int r = rowblock + v + 8 * hi;
    int g = SSTART + r;
    rlab[v] = labels_full[g];
    int gm = (g >= BBN) ? g - BBN : g;
    dv[v] = (gm < DIAGTH) ? 1.0f : 0.0f;   // self_mask diagonal value at (g,g)
  }

  // =================== pass 1: online max / masked sum-exp ===================
  float m[8], s[8];
#pragma unroll
  for (int v = 0; v < 8; ++v) { m[v] = -INFINITY; s[v] = 0.0f; }

  issue_tile(0, 0);
  for (int t = 0; t < NTILE; ++t) {
    const int cur = t & 1;
    asm volatile("s_wait_asynccnt 0x0" ::: "memory");  // own async loads landed in LDS
    __syncthreads();                                   // whole tile visible to all waves
    if (t + 1 < NTILE) issue_tile((t + 1) * 16, cur ^ 1);  // prefetch next tile

    v8f acc = {};
#pragma unroll
    for (int kk = 0; kk < 32; ++kk) {
      v2f b = *(const v2f*)&Bt[cur][lo * DN + 4 * kk + 2 * hi];
      acc = __builtin_amdgcn_wmma_f32_16x16x4_f32(false, areg[kk], false, b,
                                                  (short)0, acc, false, false);
    }

    const int clab = labels_full[t * 16 + lo];
#pragma unroll
    for (int v = 0; v < 8; ++v) {
      float x  = acc[v] * INV_T;
      float lm = (clab != rlab[v]) ? 1.0f : 0.0f;    // logits_mask
      float nm = fmaxf(m[v], x);
      s[v] = s[v] * expf(m[v] - nm) + lm * expf(x - nm);
      m[v] = nm;
    }
  }

  // combine (m,s) across the 16 lanes of each half-wave
#pragma unroll
  for (int off = 1; off < 16; off <<= 1) {
#pragma unroll
    for (int v = 0; v < 8; ++v) {
      float om = __shfl_xor(m[v], off, 32);
      float os = __shfl_xor(s[v], off, 32);
      float nm = fmaxf(m[v], om);
      s[v] = s[v] * expf(m[v] - nm) + os * expf(om - nm);
      m[v] = nm;
    }
  }
  float L[8];
#pragma unroll
  for (int v = 0; v < 8; ++v) L[v] = m[v] + logf(s[v]);   // lse (row max + log sum)

  // =================== pass 2: masked sum of min(0, x - lse) ===================
  float ps[8], ms[8];
#pragma unroll
  for (int v = 0; v < 8; ++v) { ps[v] = 0.0f; ms[v] = 0.0f; }

  issue_tile(0, 0);
  for (int t = 0; t < NTILE; ++t) {
    const int cur = t & 1;
    asm volatile("s_wait_asynccnt 0x0" ::: "memory");
    __syncthreads();
    if (t + 1 < NTILE) issue_tile((t + 1) * 16, cur ^ 1);

    v8f acc = {};
#pragma unroll
    for (int kk = 0; kk < 32; ++kk) {
      v2f b = *(const v2f*)&Bt[cur][lo * DN + 4 * kk + 2 * hi];
      acc = __builtin_amdgcn_wmma_f32_16x16x4_f32(false, areg[kk], false, b,
                                                  (short)0, acc, false, false);
    }

    const int clab = labels_full[t * 16 + lo];
    const int col  = t * 16 + lo;
#pragma unroll
    for (int v = 0; v < 8; ++v) {
      int g = SSTART + rowblock + v + 8 * hi;
      float mf = (clab == rlab[v]) ? ((col == g) ? dv[v] : 1.0f) : 0.0f;
      float x  = acc[v] * INV_T;
      ps[v] += mf * fminf(0.0f, x - L[v]);
      ms[v] += mf;
    }
  }

  // reduce across 16 lanes of each half-wave (deterministic shuffle tree)
#pragma unroll
  for (int off = 1; off < 16; off <<= 1) {
#pragma unroll
    for (int v = 0; v < 8; ++v) {
      ps[v] += __shfl_xor(ps[v], off, 32);
      ms[v] += __shfl_xor(ms[v], off, 32);
    }
  }

  if (lo == 0) {  // lanes 0 and 16 write rows v and v+8
#pragma unroll
    for (int v = 0; v < 8; ++v) {
      int r = rowblock + v + 8 * hi;
      rowres[r] = -ps[v] / ms[v];          // -(TEMP/BASE_TEMP) * mean_log_prob_pos
    }
  }
}

__global__ void finalize_kernel(const float* __restrict__ rowres, float* __restrict__ out) {
  __shared__ float sm[256];
  float acc = 0.0f;
  for (int i = threadIdx.x; i < NROWS; i += 256) acc += rowres[i];  // fixed order
  sm[threadIdx.x] = acc;
  __syncthreads();
  for (int st = 128; st > 0; st >>= 1) {
    if (threadIdx.x < st) sm[threadIdx.x] += sm[threadIdx.x + st];
    __syncthreads();
  }
  if (threadIdx.x == 0) {
    float l = sm[0] / (float)NROWS;
    out[0] = l; out[1] = l; out[2] = l;
  }
}

extern "C" void kernel_launch(void* const* d_in, const int* in_sizes, int n_in,
                              void* d_out, int out_size, void* d_ws, size_t ws_size,
                              hipStream_t stream) {
  const float* feat   = (const float*)d_in[0];   // (256,2,128) f32
  const int*   label  = (const int*)d_in[1];     // (256,) i32
  const float* fq     = (const float*)d_in[2];   // (4096,128) f32
  const float* fq2    = (const float*)d_in[3];   // (4096,128) f32
  const int*   lqueue = (const int*)d_in[4];     // (4096,) i32
  float* out = (float*)d_out;

  int*   labels_full = (int*)d_ws;                                   // 8704 ints
  float* rowres      = (float*)((char*)d_ws + TWOB * sizeof(int));   // 4352 floats

  build_labels_kernel<<<(TWOB + 255) / 256, 256, 0, stream>>>(label, lqueue, labels_full);
  supcon_main_kernel<<<NROWS / 128, 256, 0, stream>>>(feat, fq, fq2, labels_full, rowres);
  finalize_kernel<<<1, 256, 0, stream>>>(rowres, out);
}